// MSE_DQ_FK_31963146617337
// MI455X (gfx1250) — compile-verified
//
#include <hip/hip_runtime.h>

#define TT 4096
#define NB 32
#define NTOT (NB * TT)   // 131072 samples

typedef __attribute__((ext_vector_type(2))) float v2f;
typedef __attribute__((ext_vector_type(8))) float v8f;

struct Q { float w, x, y, z; };

// conj(a) * b  (quaternion product with first operand conjugated)
__device__ __forceinline__ Q conj_mul(Q a, Q b) {
    Q r;
    r.w = a.w*b.w + a.x*b.x + a.y*b.y + a.z*b.z;
    r.x = a.w*b.x - a.x*b.w - a.y*b.z + a.z*b.y;
    r.y = a.w*b.y + a.x*b.z - a.y*b.w - a.z*b.x;
    r.z = a.w*b.z - a.x*b.y + a.y*b.x - a.z*b.w;
    return r;
}

struct St {
    float gm[9];  // global rotation
    float gp[3];  // global position
    Q     q;      // this joint's (denormalized) quaternion = parent quat for next step
};

__device__ __forceinline__ void init_root(St& s, Q q0) {
    s.gm[0] = 1.f; s.gm[1] = 0.f; s.gm[2] = 0.f;
    s.gm[3] = 0.f; s.gm[4] = 1.f; s.gm[5] = 0.f;
    s.gm[6] = 0.f; s.gm[7] = 0.f; s.gm[8] = 1.f;
    s.gp[0] = 0.f; s.gp[1] = 0.f; s.gp[2] = 0.f;
    s.q = q0;
}

// Advance one joint: local rot from (parent quat, qj), normalize, to-matrix, FK update.
__device__ __forceinline__ void fk_step(St& s, Q qj, float ox, float oy, float oz) {
    Q l = conj_mul(s.q, qj);
    float n   = sqrtf(l.w*l.w + l.x*l.x + l.y*l.y + l.z*l.z);
    float inv = 1.0f / fmaxf(n, 1e-8f);
    float w = l.w*inv, x = l.x*inv, y = l.y*inv, z = l.z*inv;
    float R[9];
    R[0] = 1.f - 2.f*(y*y + z*z); R[1] = 2.f*(x*y - w*z);       R[2] = 2.f*(x*z + w*y);
    R[3] = 2.f*(x*y + w*z);       R[4] = 1.f - 2.f*(x*x + z*z); R[5] = 2.f*(y*z - w*x);
    R[6] = 2.f*(x*z - w*y);       R[7] = 2.f*(y*z + w*x);       R[8] = 1.f - 2.f*(x*x + y*y);

    // position uses PARENT gm (before update)
    float p0 = s.gm[0]*ox + s.gm[1]*oy + s.gm[2]*oz + s.gp[0];
    float p1 = s.gm[3]*ox + s.gm[4]*oy + s.gm[5]*oz + s.gp[1];
    float p2 = s.gm[6]*ox + s.gm[7]*oy + s.gm[8]*oz + s.gp[2];

    float ng[9];
#pragma unroll
    for (int r = 0; r < 3; r++)
#pragma unroll
        for (int c = 0; c < 3; c++)
            ng[3*r + c] = s.gm[3*r + 0]*R[c] + s.gm[3*r + 1]*R[3 + c] + s.gm[3*r + 2]*R[6 + c];
#pragma unroll
    for (int i = 0; i < 9; i++) s.gm[i] = ng[i];
    s.gp[0] = p0; s.gp[1] = p1; s.gp[2] = p2;
    s.q = qj;
}

// Streaming (read-once) load of one denormalized quaternion: 4 channels, stride TT floats.
// Non-temporal hint: ~128 MB streamed once; keep mean/std/offsets in cache instead.
__device__ __forceinline__ Q load_q(const float* __restrict__ base, int ch,
                                    const float sd[4], const float mn[4]) {
    Q q;
    q.w = __builtin_nontemporal_load(base + (size_t)(ch + 0) * TT) * sd[0] + mn[0];
    q.x = __builtin_nontemporal_load(base + (size_t)(ch + 1) * TT) * sd[1] + mn[1];
    q.y = __builtin_nontemporal_load(base + (size_t)(ch + 2) * TT) * sd[2] + mn[2];
    q.z = __builtin_nontemporal_load(base + (size_t)(ch + 3) * TT) * sd[3] + mn[3];
    return q;
}

__global__ __launch_bounds__(256)
void fk_loss_kernel(const float* __restrict__ ikp,  const float* __restrict__ decp,
                    const float* __restrict__ tgtp, const float* __restrict__ mean,
                    const float* __restrict__ stdv, const float* __restrict__ offs,
                    float* __restrict__ partials) {
    const int gid = blockIdx.x * blockDim.x + threadIdx.x;   // 0..NTOT-1
    const int b   = gid >> 12;          // / 4096
    const int t   = gid & (TT - 1);

    // ik stream: dqs = concat(zeros(8ch), input_ik) -> channel c maps to input_ik channel c-8
    const float* ib = ikp  + (size_t)b * (168 * TT) + t;
    const float* db = decp + (size_t)b * (176 * TT) + t;
    const float* gb = tgtp + (size_t)b * (176 * TT) + t;

    // target root quaternion (set_root=False for target stream)
    Q q0t;
    {
        q0t.w = __builtin_nontemporal_load(gb + (size_t)0 * TT) * stdv[0] + mean[0];
        q0t.x = __builtin_nontemporal_load(gb + (size_t)1 * TT) * stdv[1] + mean[1];
        q0t.y = __builtin_nontemporal_load(gb + (size_t)2 * TT) * stdv[2] + mean[2];
        q0t.z = __builtin_nontemporal_load(gb + (size_t)3 * TT) * stdv[3] + mean[3];
    }
    const Q qid = {1.f, 0.f, 0.f, 0.f};  // ik/dec roots forced to identity (set_root=True)

    // MSE weights folded per-term (exact: each MSE divides by its own element count)
    const float w_ee_pos  = 1.0f / ((float)NTOT * 15.f);    // 5 EE joints * 3
    const float w_ee_mat  = 1.0f / ((float)NTOT * 45.f);    // 5 EE joints * 9
    const float w_reg_pos = 0.1f / ((float)NTOT * 48.f);    // 16 joints * 3
    const float w_reg_mat = 0.1f / ((float)NTOT * 144.f);   // 16 joints * 9

    St sik, sdec, stgt;
    St svik, svdec, svtgt;  // state saved at joint 11 (branch point)
    float acc = 0.f;

    // DFS over the tree: chains 1-4, 5-8, 9-13, then 14-17 and 18-21 off joint 11.
    const int order_[21] = {1,2,3,4, 5,6,7,8, 9,10,11,12,13, 14,15,16,17, 18,19,20,21};
#pragma unroll
    for (int i = 0; i < 21; i++) {
        const int j = order_[i];
        if (j == 1 || j == 5 || j == 9) {           // chain starts at root
            init_root(sik, qid); init_root(sdec, qid); init_root(stgt, q0t);
        }
        if (j == 14 || j == 18) {                   // branch restarts at joint 11
            sik = svik; sdec = svdec; stgt = svtgt;
        }

        const float ox = offs[3*j + 0], oy = offs[3*j + 1], oz = offs[3*j + 2];
        float mn[4], sd[4];
#pragma unroll
        for (int k = 0; k < 4; k++) { mn[k] = mean[j*8 + k]; sd[k] = stdv[j*8 + k]; }

        Q qi = load_q(ib, j*8 - 8, sd, mn);
        Q qd = load_q(db, j*8,     sd, mn);
        Q qg = load_q(gb, j*8,     sd, mn);

        fk_step(sik,  qi, ox, oy, oz);
        fk_step(sdec, qd, ox, oy, oz);
        fk_step(stgt, qg, ox, oy, oz);

        const bool is_ee = (j == 4) || (j == 8) || (j == 13) || (j == 17) || (j == 21);
        if (is_ee) {
            float sp = 0.f, sm = 0.f;
#pragma unroll
            for (int k = 0; k < 3; k++) { float d = sik.gp[k] - stgt.gp[k]; sp += d*d; }
#pragma unroll
            for (int k = 0; k < 9; k++) { float d = sik.gm[k] - stgt.gm[k]; sm += d*d; }
            acc += w_ee_pos * sp + w_ee_mat * sm;
        } else {
            float sp = 0.f, sm = 0.f;
#pragma unroll
            for (int k = 0; k < 3; k++) { float d = sdec.gp[k] - sik.gp[k]; sp += d*d; }
#pragma unroll
            for (int k = 0; k < 9; k++) { float d = sdec.gm[k] - sik.gm[k]; sm += d*d; }
            acc += w_reg_pos * sp + w_reg_mat * sm;
        }

        if (j == 11) { svik = sik; svdec = sdec; svtgt = stgt; }
    }

    // wave32 shuffle reduction (deterministic fixed order)
#pragma unroll
    for (int o = 16; o > 0; o >>= 1) acc += __shfl_down(acc, o, 32);

    __shared__ float smem[8];
    const int lane = threadIdx.x & 31;
    const int wv   = threadIdx.x >> 5;
    if (lane == 0) smem[wv] = acc;
    __syncthreads();
    if (threadIdx.x == 0) {
        float s = 0.f;
#pragma unroll
        for (int w2 = 0; w2 < 8; w2++) s += smem[w2];
        partials[blockIdx.x] = s;
    }
}

// Deterministic final reduction over n (multiple of 64) block partials, one full wave.
//
// Uses V_WMMA_F32_16X16X4_F32 in a lane-layout-independent way:
//   A (16x4 f32, 2 VGPRs/lane) <- 64 per-lane partial sums (any bijective packing)
//   B (4x16 f32)               <- all-ones (all-ones under ANY layout)
//   => D[m][n] = sum_k A[m][k]  for every n, so sum_all(D) = 16 * sum_all(A).
// Summing every lane's 8 D registers, wave-reducing, and scaling by exact 1/16
// yields sum(partials) irrespective of the hardware's VGPR<->matrix mapping.
// EXEC is all-ones (single 32-thread wave, uniform control flow) as WMMA requires.
__global__ void fk_loss_finalize(const float* __restrict__ partials,
                                 float* __restrict__ out, int n) {
    const int lane = (int)threadIdx.x;   // 0..31
    float s0 = 0.f, s1 = 0.f;
    for (int m = 0; m < n; m += 64) {    // fixed order -> deterministic
        s0 += partials[m + lane];
        s1 += partials[m + lane + 32];
    }
    v2f a;  a.x = s0;  a.y = s1;
    v2f bones; bones.x = 1.0f; bones.y = 1.0f;
    v8f c = {};
    v8f d = __builtin_amdgcn_wmma_f32_16x16x4_f32(
        /*neg_a=*/false, a, /*neg_b=*/false, bones,
        /*c_mod=*/(short)0, c, /*reuse_a=*/false, /*reuse_b=*/false);

    float s = d[0] + d[1] + d[2] + d[3] + d[4] + d[5] + d[6] + d[7];
#pragma unroll
    for (int o = 16; o > 0; o >>= 1) s += __shfl_down(s, o, 32);
    if (lane == 0) out[0] = s * (1.0f / 16.0f);  // exact power-of-two scale
}

extern "C" void kernel_launch(void* const* d_in, const int* in_sizes, int n_in,
                              void* d_out, int out_size, void* d_ws, size_t ws_size,
                              hipStream_t stream) {
    // setup_inputs order: 0=input(unused), 1=input_ik, 2=input_decoder, 3=target,
    //                     4=mean_dqs, 5=std_dqs, 6=offsets
    const float* ikp  = (const float*)d_in[1];
    const float* decp = (const float*)d_in[2];
    const float* tgtp = (const float*)d_in[3];
    const float* mean = (const float*)d_in[4];
    const float* stdv = (const float*)d_in[5];
    const float* offs = (const float*)d_in[6];

    float* partials = (float*)d_ws;          // 512 floats of scratch
    const int threads = 256;
    const int blocks  = NTOT / threads;      // 512 blocks (multiple of 64), one thread per (b,t)

    fk_loss_kernel<<<blocks, threads, 0, stream>>>(ikp, decp, tgtp, mean, stdv, offs, partials);
    fk_loss_finalize<<<1, 32, 0, stream>>>(partials, (float*)d_out, blocks);
}